// SoftPhongShader_7645041787393
// MI455X (gfx1250) — compile-verified
//
#include <hip/hip_runtime.h>
#include <hip/hip_bf16.h>

#define SIGMA_F   1e-5f
#define GAMMA_F   1e-4f
#define ZNEAR_F   1.0f
#define ZFAR_F    100.0f
#define EPS_F     1e-10f

typedef __attribute__((ext_vector_type(4))) unsigned int v4u;
typedef __attribute__((ext_vector_type(8))) int          v8i;
typedef __attribute__((ext_vector_type(4))) int          v4i;

// ---------------------------------------------------------------------------
// Kernel 1: repack verts / verts_normals / verts_rgb into an interleaved
// [V][12] float table (48B/vertex, 16B aligned) in d_ws so each vertex gather
// is 3 contiguous float4 loads (~1 L2 line) instead of 3 scattered 12B reads.
// ---------------------------------------------------------------------------
__global__ __launch_bounds__(256)
void pack_vertex_attrs(const float* __restrict__ verts,
                       const float* __restrict__ vnorm,
                       const float* __restrict__ vrgb,
                       float4* __restrict__ packed, int V) {
    int v = blockIdx.x * blockDim.x + threadIdx.x;
    if (v >= V) return;
    float px = verts[3*v+0], py = verts[3*v+1], pz = verts[3*v+2];
    float nx = vnorm[3*v+0], ny = vnorm[3*v+1], nz = vnorm[3*v+2];
    float tr = vrgb [3*v+0], tg = vrgb [3*v+1], tb = vrgb [3*v+2];
    packed[3*v+0] = make_float4(px, py, pz, nx);
    packed[3*v+1] = make_float4(ny, nz, tr, tg);
    packed[3*v+2] = make_float4(tb, 0.f, 0.f, 0.f);
}

__device__ __forceinline__ void vattr_packed(const float4* __restrict__ pk, int v,
    float& px, float& py, float& pz, float& nx, float& ny, float& nz,
    float& tr, float& tg, float& tb) {
    float4 a0 = pk[3*v+0];
    float4 a1 = pk[3*v+1];
    float4 a2 = pk[3*v+2];
    px = a0.x; py = a0.y; pz = a0.z;
    nx = a0.w; ny = a1.x; nz = a1.y;
    tr = a1.z; tg = a1.w; tb = a2.x;
}

__device__ __forceinline__ void vattr_raw(const float* __restrict__ V,
    const float* __restrict__ N, const float* __restrict__ C, int v,
    float& px, float& py, float& pz, float& nx, float& ny, float& nz,
    float& tr, float& tg, float& tb) {
    px = V[3*v+0]; py = V[3*v+1]; pz = V[3*v+2];
    nx = N[3*v+0]; ny = N[3*v+1]; nz = N[3*v+2];
    tr = C[3*v+0]; tg = C[3*v+1]; tb = C[3*v+2];
}

struct ShadeUni {
    float Lx, Ly, Lz, Cx, Cy, Cz;
    float kaR, kaG, kaB, kdR, kdG, kdB, ksR, ksG, ksB;
};

__device__ __forceinline__ ShadeUni load_uniforms(
    const float* __restrict__ lloc, const float* __restrict__ cam,
    const float* __restrict__ la, const float* __restrict__ ld,
    const float* __restrict__ ls, const float* __restrict__ ma,
    const float* __restrict__ md, const float* __restrict__ ms) {
    ShadeUni u;
    u.Lx = lloc[0]; u.Ly = lloc[1]; u.Lz = lloc[2];
    u.Cx = cam[0];  u.Cy = cam[1];  u.Cz = cam[2];
    u.kaR = ma[0]*la[0]; u.kaG = ma[1]*la[1]; u.kaB = ma[2]*la[2];
    u.kdR = md[0]*ld[0]; u.kdG = md[1]*ld[1]; u.kdB = md[2]*ld[2];
    u.ksR = ms[0]*ls[0]; u.ksG = ms[1]*ls[1]; u.ksB = ms[2]*ls[2];
    return u;
}

// ---------------------------------------------------------------------------
// Per-pixel Phong + softmax blend on register-resident fragment data.
// ---------------------------------------------------------------------------
template<bool PACKED>
__device__ __forceinline__ float4 shade_pixel(
    const int fk[8], const float zk[8], const float dk[8], const float bf[24],
    const ShadeUni& u,
    const int*    __restrict__ faces,
    const float4* __restrict__ packed,
    const float*  __restrict__ verts,
    const float*  __restrict__ vnorm,
    const float*  __restrict__ vrgb) {

    // gfx1250: prefetch the faces rows for all 8 fragments up front
    // (global_prefetch_b8) so the dependent vertex gathers start sooner.
    #pragma unroll
    for (int k = 0; k < 8; ++k) {
        int f = fk[k] < 0 ? 0 : fk[k];
        __builtin_prefetch(&faces[3*f], 0, 3);
    }

    float cr[8], cg[8], cb[8], pr[8], zi[8];

    #pragma unroll
    for (int k = 0; k < 8; ++k) {
        int face = fk[k];
        float r = 0.f, g = 0.f, b = 0.f, prob = 0.f, zv = 0.f;
        if (face >= 0) {
            int i0 = faces[3*face+0], i1 = faces[3*face+1], i2 = faces[3*face+2];
            float p0x,p0y,p0z,n0x,n0y,n0z,t0r,t0g,t0b;
            float p1x,p1y,p1z,n1x,n1y,n1z,t1r,t1g,t1b;
            float p2x,p2y,p2z,n2x,n2y,n2z,t2r,t2g,t2b;
            if (PACKED) {
                vattr_packed(packed, i0, p0x,p0y,p0z,n0x,n0y,n0z,t0r,t0g,t0b);
                vattr_packed(packed, i1, p1x,p1y,p1z,n1x,n1y,n1z,t1r,t1g,t1b);
                vattr_packed(packed, i2, p2x,p2y,p2z,n2x,n2y,n2z,t2r,t2g,t2b);
            } else {
                vattr_raw(verts, vnorm, vrgb, i0, p0x,p0y,p0z,n0x,n0y,n0z,t0r,t0g,t0b);
                vattr_raw(verts, vnorm, vrgb, i1, p1x,p1y,p1z,n1x,n1y,n1z,t1r,t1g,t1b);
                vattr_raw(verts, vnorm, vrgb, i2, p2x,p2y,p2z,n2x,n2y,n2z,t2r,t2g,t2b);
            }
            float w0 = bf[3*k+0], w1 = bf[3*k+1], w2 = bf[3*k+2];
            float px = fmaf(w0,p0x, fmaf(w1,p1x, w2*p2x));
            float py = fmaf(w0,p0y, fmaf(w1,p1y, w2*p2y));
            float pz = fmaf(w0,p0z, fmaf(w1,p1z, w2*p2z));
            float nx = fmaf(w0,n0x, fmaf(w1,n1x, w2*n2x));
            float ny = fmaf(w0,n0y, fmaf(w1,n1y, w2*n2y));
            float nz = fmaf(w0,n0z, fmaf(w1,n1z, w2*n2z));
            float tr = fmaf(w0,t0r, fmaf(w1,t1r, w2*t2r));
            float tg = fmaf(w0,t0g, fmaf(w1,t1g, w2*t2g));
            float tb = fmaf(w0,t0b, fmaf(w1,t1b, w2*t2b));
            float ninv = rsqrtf(fmaxf(nx*nx + ny*ny + nz*nz, 1e-12f));
            nx *= ninv; ny *= ninv; nz *= ninv;
            float lx = u.Lx - px, ly = u.Ly - py, lz = u.Lz - pz;
            float linv = rsqrtf(fmaxf(lx*lx + ly*ly + lz*lz, 1e-12f));
            lx *= linv; ly *= linv; lz *= linv;
            float cosA = nx*lx + ny*ly + nz*lz;
            float cpos = fmaxf(cosA, 0.f);
            float vx = u.Cx - px, vy = u.Cy - py, vz = u.Cz - pz;
            float vinv = rsqrtf(fmaxf(vx*vx + vy*vy + vz*vz, 1e-12f));
            vx *= vinv; vy *= vinv; vz *= vinv;
            float twoc = 2.f * cosA;
            float rx = fmaf(twoc, nx, -lx);
            float ry = fmaf(twoc, ny, -ly);
            float rz = fmaf(twoc, nz, -lz);
            float al = fmaxf(vx*rx + vy*ry + vz*rz, 0.f);
            al = (cosA > 0.f) ? al : 0.f;
            float a2 = al*al, a4 = a2*a2, a8 = a4*a4;
            float a16 = a8*a8, a32 = a16*a16, a64 = a32*a32;
            r = fmaf(fmaf(u.kdR, cpos, u.kaR), tr, u.ksR * a64);
            g = fmaf(fmaf(u.kdG, cpos, u.kaG), tg, u.ksG * a64);
            b = fmaf(fmaf(u.kdB, cpos, u.kaB), tb, u.ksB * a64);
            prob = 1.f / (1.f + __expf(dk[k] * (1.f/SIGMA_F)));
            zv = (ZFAR_F - zk[k]) * (1.f / (ZFAR_F - ZNEAR_F));
        }
        cr[k] = r; cg[k] = g; cb[k] = b; pr[k] = prob; zi[k] = zv;
    }

    float zmax = EPS_F;
    #pragma unroll
    for (int k = 0; k < 8; ++k) zmax = fmaxf(zmax, zi[k]);
    float keep = 1.f;
    #pragma unroll
    for (int k = 0; k < 8; ++k) keep *= (1.f - pr[k]);
    float alpha = 1.f - keep;

    const float invg = 1.f / GAMMA_F;
    float delta = fmaxf(__expf((EPS_F - zmax) * invg), EPS_F);
    float denom = delta, sr = 0.f, sg = 0.f, sb = 0.f;
    #pragma unroll
    for (int k = 0; k < 8; ++k) {
        float w = pr[k] * __expf((zi[k] - zmax) * invg);
        denom += w;
        sr = fmaf(w, cr[k], sr);
        sg = fmaf(w, cg[k], sg);
        sb = fmaf(w, cb[k], sb);
    }
    float invd = 1.f / denom;       // BG=(0,0,0): delta term adds nothing to rgb
    return make_float4(sr*invd, sg*invd, sb*invd, alpha);
}

// ---------------------------------------------------------------------------
// Tensor Data Mover: 1-row 2D tile descriptor copying `nelem` contiguous
// 4-byte elements from global memory to LDS (ISA cdna5 §8: D# groups 0/1;
// groups 2/3 zero => tile_dim2/3 unused). Issued once per wave, EXEC ignored.
// ---------------------------------------------------------------------------
__device__ __forceinline__ void tdm_load_1d(unsigned lds_byte_addr,
                                            const void* gsrc, unsigned nelem) {
    unsigned long long ga = (unsigned long long)(uintptr_t)gsrc;
    v4u g0;
    g0.x = 1u;                                           // count=1, is_restore=0
    g0.y = lds_byte_addr;                                // lds_addr
    g0.z = (unsigned)(ga & 0xffffffffull);               // global_addr[31:0]
    g0.w = (unsigned)((ga >> 32) & 0x01ffffffull)        // global_addr[56:32]
         | (2u << 30);                                   // type=2 ("image")
    v8i g1;
    g1[0] = 0x00020000;                                  // data_size=2 (4B), mask=0
    g1[1] = (int)((nelem & 0xffffu) << 16);              // tensor_dim0[15:0]
    g1[2] = (int)(((nelem >> 16) & 0xffffu) | (1u<<16)); // tensor_dim0[31:16], tensor_dim1=1
    g1[3] = (int)(nelem << 16);                          // tile_dim0 = nelem (<=65535)
    g1[4] = 1;                                           // tile_dim1=1, tile_dim2=0
    g1[5] = (int)nelem;                                  // tensor_dim0_stride[31:0]
    g1[6] = 0;                                           // stride hi / dim1_stride lo
    g1[7] = 0;
    v4i z4 = {};
#if defined(__clang_major__) && (__clang_major__ >= 23)
    v8i z8 = {};
    __builtin_amdgcn_tensor_load_to_lds(g0, g1, z4, z4, z8, 0);
#else
    __builtin_amdgcn_tensor_load_to_lds(g0, g1, z4, z4, 0);
#endif
}

// LDS staging layout for one 256-pixel block
#define OFF_PTF   0u
#define OFF_Z     8192u
#define OFF_D     16384u
#define OFF_BARY  24576u
#define SMEM_SZ   49152u

// ---------------------------------------------------------------------------
// Main shader, TDM variant: 4 waves each DMA one slab (ptf/z/d/bary) of the
// block's 256 pixels into LDS, wait on TENSORcnt, barrier, then all threads
// consume via ds_load_b128 and keep the VMEM pipe free for the random gathers.
// Requires a full block (handled by launcher).
// ---------------------------------------------------------------------------
template<bool PACKED>
__global__ __launch_bounds__(256)
void soft_phong_tdm(const int*    __restrict__ ptf,
                    const float*  __restrict__ bary,
                    const float*  __restrict__ zbuf,
                    const float*  __restrict__ dists,
                    const int*    __restrict__ faces,
                    const float4* __restrict__ packed,
                    const float*  __restrict__ verts,
                    const float*  __restrict__ vnorm,
                    const float*  __restrict__ vrgb,
                    const float*  __restrict__ lloc,
                    const float*  __restrict__ la,
                    const float*  __restrict__ ld,
                    const float*  __restrict__ ls,
                    const float*  __restrict__ ma,
                    const float*  __restrict__ md,
                    const float*  __restrict__ ms,
                    const float*  __restrict__ cam,
                    float4*       __restrict__ out) {
    __shared__ __align__(16) unsigned char smem[SMEM_SZ];
    const int tid = threadIdx.x;
    const int blockBase = blockIdx.x * 256;     // first pixel of this block
    const int p = blockBase + tid;

    // wave-uniform branch: waves 0..3 each issue one TDM descriptor
    // (TDM ignores EXEC, so the guard must be wave-granular).
    const int wv = tid >> 5;
    if (wv < 4) {
        const void* src;
        unsigned nelem, ldsoff;
        switch (wv) {
        case 0:  src = ptf   + (size_t)blockBase * 8;  nelem = 2048; ldsoff = OFF_PTF;  break;
        case 1:  src = zbuf  + (size_t)blockBase * 8;  nelem = 2048; ldsoff = OFF_Z;    break;
        case 2:  src = dists + (size_t)blockBase * 8;  nelem = 2048; ldsoff = OFF_D;    break;
        default: src = bary  + (size_t)blockBase * 24; nelem = 6144; ldsoff = OFF_BARY; break;
        }
        tdm_load_1d(ldsoff, src, nelem);
        __builtin_amdgcn_s_wait_tensorcnt(0);
    }
    __syncthreads();

    // consume staged slabs from LDS (ds_load_b128)
    const int4*   sptf = (const int4*)  (smem + OFF_PTF);
    const float4* sz   = (const float4*)(smem + OFF_Z);
    const float4* sd   = (const float4*)(smem + OFF_D);
    const float4* sb   = (const float4*)(smem + OFF_BARY);

    int4 fA = sptf[tid*2+0], fB = sptf[tid*2+1];
    int fk[8] = { fA.x, fA.y, fA.z, fA.w, fB.x, fB.y, fB.z, fB.w };
    float4 zA = sz[tid*2+0], zB = sz[tid*2+1];
    float zk[8] = { zA.x, zA.y, zA.z, zA.w, zB.x, zB.y, zB.z, zB.w };
    float4 dA = sd[tid*2+0], dB = sd[tid*2+1];
    float dk[8] = { dA.x, dA.y, dA.z, dA.w, dB.x, dB.y, dB.z, dB.w };
    float4 q0 = sb[tid*6+0], q1 = sb[tid*6+1], q2 = sb[tid*6+2];
    float4 q3 = sb[tid*6+3], q4 = sb[tid*6+4], q5 = sb[tid*6+5];
    float bf[24] = { q0.x,q0.y,q0.z,q0.w, q1.x,q1.y,q1.z,q1.w,
                     q2.x,q2.y,q2.z,q2.w, q3.x,q3.y,q3.z,q3.w,
                     q4.x,q4.y,q4.z,q4.w, q5.x,q5.y,q5.z,q5.w };

    ShadeUni u = load_uniforms(lloc, cam, la, ld, ls, ma, md, ms);
    out[p] = shade_pixel<PACKED>(fk, zk, dk, bf, u, faces, packed,
                                 verts, vnorm, vrgb);
}

// ---------------------------------------------------------------------------
// Direct-load variant (tail blocks / fallback): vectorized global loads.
// ---------------------------------------------------------------------------
template<bool PACKED>
__global__ __launch_bounds__(256)
void soft_phong_direct(const int*    __restrict__ ptf,
                       const float*  __restrict__ bary,
                       const float*  __restrict__ zbuf,
                       const float*  __restrict__ dists,
                       const int*    __restrict__ faces,
                       const float4* __restrict__ packed,
                       const float*  __restrict__ verts,
                       const float*  __restrict__ vnorm,
                       const float*  __restrict__ vrgb,
                       const float*  __restrict__ lloc,
                       const float*  __restrict__ la,
                       const float*  __restrict__ ld,
                       const float*  __restrict__ ls,
                       const float*  __restrict__ ma,
                       const float*  __restrict__ md,
                       const float*  __restrict__ ms,
                       const float*  __restrict__ cam,
                       float4*       __restrict__ out,
                       int pix_offset, int npix) {
    int p = pix_offset + blockIdx.x * blockDim.x + threadIdx.x;
    if (p >= npix) return;

    const int4* ptf4 = (const int4*)ptf + (size_t)p*2;
    int4 fA = ptf4[0], fB = ptf4[1];
    int fk[8] = { fA.x, fA.y, fA.z, fA.w, fB.x, fB.y, fB.z, fB.w };
    const float4* z4 = (const float4*)zbuf + (size_t)p*2;
    float4 zA = z4[0], zB = z4[1];
    float zk[8] = { zA.x, zA.y, zA.z, zA.w, zB.x, zB.y, zB.z, zB.w };
    const float4* d4 = (const float4*)dists + (size_t)p*2;
    float4 dA = d4[0], dB = d4[1];
    float dk[8] = { dA.x, dA.y, dA.z, dA.w, dB.x, dB.y, dB.z, dB.w };
    const float4* b4 = (const float4*)bary + (size_t)p*6;
    float4 q0 = b4[0], q1 = b4[1], q2 = b4[2], q3 = b4[3], q4 = b4[4], q5 = b4[5];
    float bf[24] = { q0.x,q0.y,q0.z,q0.w, q1.x,q1.y,q1.z,q1.w,
                     q2.x,q2.y,q2.z,q2.w, q3.x,q3.y,q3.z,q3.w,
                     q4.x,q4.y,q4.z,q4.w, q5.x,q5.y,q5.z,q5.w };

    ShadeUni u = load_uniforms(lloc, cam, la, ld, ls, ma, md, ms);
    out[p] = shade_pixel<PACKED>(fk, zk, dk, bf, u, faces, packed,
                                 verts, vnorm, vrgb);
}

extern "C" void kernel_launch(void* const* d_in, const int* in_sizes, int n_in,
                              void* d_out, int out_size, void* d_ws, size_t ws_size,
                              hipStream_t stream) {
    const int*   ptf   = (const int*)  d_in[0];
    const float* bary  = (const float*)d_in[1];
    const float* zbuf  = (const float*)d_in[2];
    const float* dists = (const float*)d_in[3];
    const float* verts = (const float*)d_in[4];
    const int*   faces = (const int*)  d_in[5];
    const float* vnorm = (const float*)d_in[6];
    const float* vrgb  = (const float*)d_in[7];
    const float* lloc  = (const float*)d_in[8];
    const float* la    = (const float*)d_in[9];
    const float* ld    = (const float*)d_in[10];
    const float* ls    = (const float*)d_in[11];
    const float* ma    = (const float*)d_in[12];
    const float* md    = (const float*)d_in[13];
    const float* ms    = (const float*)d_in[14];
    const float* cam   = (const float*)d_in[15];

    const int K    = 8;
    const int npix = in_sizes[0] / K;          // N*H*W
    const int V    = in_sizes[4] / 3;

    size_t packed_bytes = (size_t)V * 12 * sizeof(float);
    bool use_packed = (ws_size >= packed_bytes);
    float4* packed = (float4*)d_ws;

    const int nfull = npix / 256;              // full 256-pixel blocks -> TDM
    const int rem   = npix - nfull * 256;      // tail -> direct loads

    if (use_packed) {
        pack_vertex_attrs<<<(V + 255) / 256, 256, 0, stream>>>(
            verts, vnorm, vrgb, packed, V);
        if (nfull > 0)
            soft_phong_tdm<true><<<nfull, 256, 0, stream>>>(
                ptf, bary, zbuf, dists, faces, packed, verts, vnorm, vrgb,
                lloc, la, ld, ls, ma, md, ms, cam, (float4*)d_out);
        if (rem > 0)
            soft_phong_direct<true><<<1, 256, 0, stream>>>(
                ptf, bary, zbuf, dists, faces, packed, verts, vnorm, vrgb,
                lloc, la, ld, ls, ma, md, ms, cam, (float4*)d_out,
                nfull * 256, npix);
    } else {
        if (nfull > 0)
            soft_phong_tdm<false><<<nfull, 256, 0, stream>>>(
                ptf, bary, zbuf, dists, faces, packed, verts, vnorm, vrgb,
                lloc, la, ld, ls, ma, md, ms, cam, (float4*)d_out);
        if (rem > 0)
            soft_phong_direct<false><<<1, 256, 0, stream>>>(
                ptf, bary, zbuf, dists, faces, packed, verts, vnorm, vrgb,
                lloc, la, ld, ls, ma, md, ms, cam, (float4*)d_out,
                nfull * 256, npix);
    }
}